// GatingFunction_27814208208930
// MI455X (gfx1250) — compile-verified
//
#include <hip/hip_runtime.h>

#define INPUT_SIZE  2048
#define HIDDEN_SIZE 1024
#define NUM_EXPERTS 64
#define BATCH       8192
#define MARGIN      0.1f

typedef __attribute__((ext_vector_type(2))) float v2f;
typedef __attribute__((ext_vector_type(4))) float v4f;
typedef __attribute__((ext_vector_type(8))) float v8f;

// ---------------------------------------------------------------------------
// Kernel A: h = relu(x @ W1^T + b1)
// block = 256 threads (8 wave32). Block tile: M=256, N=64, K-step=32.
// Each wave: 32 rows x 64 cols -> 8 f32 accumulators, 6 LDS b64 loads per
// 8 x V_WMMA_F32_16X16X4_F32 (register blocking to cut LDS pressure).
// ---------------------------------------------------------------------------
__global__ __launch_bounds__(256) void gemm1_relu_kernel(
    const float* __restrict__ x, const float* __restrict__ W1,
    const float* __restrict__ b1, float* __restrict__ h)
{
    __shared__ float sX[256][36];   // row stride 144B (16B aligned), 36.9 KB
    __shared__ float sW[64][36];    // 9.2 KB

    const int tid   = threadIdx.x;
    const int wave  = tid >> 5;
    const int lane  = tid & 31;
    const int m0    = blockIdx.y * 256;
    const int n0    = blockIdx.x * 64;
    const int am    = lane & 15;          // M (A) / N (B) index within 16-tile
    const int khalf = (lane >> 4) << 1;   // lanes 16-31 hold K+2,K+3

    v8f c00 = {}, c01 = {}, c02 = {}, c03 = {};
    v8f c10 = {}, c11 = {}, c12 = {}, c13 = {};

    for (int k0 = 0; k0 < INPUT_SIZE; k0 += 32) {
        // cooperative, fully-coalesced float4 staging
        {
            const int c4 = (tid & 7) << 2;        // 8 threads per 32-float row
            const int rr = tid >> 3;              // 0..31
            #pragma unroll
            for (int r = 0; r < 256; r += 32) {
                v4f v = *(const v4f*)(x + (size_t)(m0 + r + rr) * INPUT_SIZE + k0 + c4);
                *(v4f*)&sX[r + rr][c4] = v;
            }
            #pragma unroll
            for (int r = 0; r < 64; r += 32) {
                v4f v = *(const v4f*)(W1 + (size_t)(n0 + r + rr) * INPUT_SIZE + k0 + c4);
                *(v4f*)&sW[r + rr][c4] = v;
            }
        }
        __syncthreads();

        #pragma unroll
        for (int kk = 0; kk < 32; kk += 4) {
            v2f a0 = *(const v2f*)&sX[(wave << 5) + am     ][kk + khalf];
            v2f a1 = *(const v2f*)&sX[(wave << 5) + am + 16][kk + khalf];
            v2f b0 = *(const v2f*)&sW[ 0 + am][kk + khalf];
            v2f b1v= *(const v2f*)&sW[16 + am][kk + khalf];
            v2f b2v= *(const v2f*)&sW[32 + am][kk + khalf];
            v2f b3v= *(const v2f*)&sW[48 + am][kk + khalf];
            c00 = __builtin_amdgcn_wmma_f32_16x16x4_f32(false, a0, false, b0,  (short)0, c00, false, false);
            c01 = __builtin_amdgcn_wmma_f32_16x16x4_f32(false, a0, false, b1v, (short)0, c01, false, false);
            c02 = __builtin_amdgcn_wmma_f32_16x16x4_f32(false, a0, false, b2v, (short)0, c02, false, false);
            c03 = __builtin_amdgcn_wmma_f32_16x16x4_f32(false, a0, false, b3v, (short)0, c03, false, false);
            c10 = __builtin_amdgcn_wmma_f32_16x16x4_f32(false, a1, false, b0,  (short)0, c10, false, false);
            c11 = __builtin_amdgcn_wmma_f32_16x16x4_f32(false, a1, false, b1v, (short)0, c11, false, false);
            c12 = __builtin_amdgcn_wmma_f32_16x16x4_f32(false, a1, false, b2v, (short)0, c12, false, false);
            c13 = __builtin_amdgcn_wmma_f32_16x16x4_f32(false, a1, false, b3v, (short)0, c13, false, false);
        }
        __syncthreads();
    }

    // epilogue: bias + relu + store (two 16-row fragments per wave)
    const int colb = n0 + am;
    const float bb0 = b1[colb], bb1 = b1[colb + 16], bb2 = b1[colb + 32], bb3 = b1[colb + 48];
    const int rhalf = (lane >> 4) << 3;
    #pragma unroll
    for (int i = 0; i < 8; ++i) {
        size_t ro = (size_t)(m0 + (wave << 5) + i + rhalf) * HIDDEN_SIZE;
        h[ro + colb     ] = fmaxf(c00[i] + bb0, 0.0f);
        h[ro + colb + 16] = fmaxf(c01[i] + bb1, 0.0f);
        h[ro + colb + 32] = fmaxf(c02[i] + bb2, 0.0f);
        h[ro + colb + 48] = fmaxf(c03[i] + bb3, 0.0f);
        ro += (size_t)16 * HIDDEN_SIZE;
        h[ro + colb     ] = fmaxf(c10[i] + bb0, 0.0f);
        h[ro + colb + 16] = fmaxf(c11[i] + bb1, 0.0f);
        h[ro + colb + 32] = fmaxf(c12[i] + bb2, 0.0f);
        h[ro + colb + 48] = fmaxf(c13[i] + bb3, 0.0f);
    }
}

// ---------------------------------------------------------------------------
// Kernel B: probs = softmax(h @ W2^T + b2) over experts
// block = 256 threads (8 waves). Tile M=256, N=64 (all experts), K-step=32.
// Each wave: 32 rows x 64 experts, softmax in registers with half-wave
// (16-lane) shuffle reductions (xor 8/4/2/1 stays inside each half).
// ---------------------------------------------------------------------------
__global__ __launch_bounds__(256) void gemm2_softmax_kernel(
    const float* __restrict__ h, const float* __restrict__ W2,
    const float* __restrict__ b2, float* __restrict__ probs)
{
    __shared__ float sH[256][36];
    __shared__ float sW[64][36];

    const int tid   = threadIdx.x;
    const int wave  = tid >> 5;
    const int lane  = tid & 31;
    const int m0    = blockIdx.x * 256;
    const int am    = lane & 15;
    const int khalf = (lane >> 4) << 1;

    v8f c00 = {}, c01 = {}, c02 = {}, c03 = {};
    v8f c10 = {}, c11 = {}, c12 = {}, c13 = {};

    for (int k0 = 0; k0 < HIDDEN_SIZE; k0 += 32) {
        {
            const int c4 = (tid & 7) << 2;
            const int rr = tid >> 3;
            #pragma unroll
            for (int r = 0; r < 256; r += 32) {
                v4f v = *(const v4f*)(h + (size_t)(m0 + r + rr) * HIDDEN_SIZE + k0 + c4);
                *(v4f*)&sH[r + rr][c4] = v;
            }
            #pragma unroll
            for (int r = 0; r < 64; r += 32) {
                v4f v = *(const v4f*)(W2 + (size_t)(r + rr) * HIDDEN_SIZE + k0 + c4);
                *(v4f*)&sW[r + rr][c4] = v;
            }
        }
        __syncthreads();

        #pragma unroll
        for (int kk = 0; kk < 32; kk += 4) {
            v2f a0 = *(const v2f*)&sH[(wave << 5) + am     ][kk + khalf];
            v2f a1 = *(const v2f*)&sH[(wave << 5) + am + 16][kk + khalf];
            v2f b0 = *(const v2f*)&sW[ 0 + am][kk + khalf];
            v2f b1v= *(const v2f*)&sW[16 + am][kk + khalf];
            v2f b2v= *(const v2f*)&sW[32 + am][kk + khalf];
            v2f b3v= *(const v2f*)&sW[48 + am][kk + khalf];
            c00 = __builtin_amdgcn_wmma_f32_16x16x4_f32(false, a0, false, b0,  (short)0, c00, false, false);
            c01 = __builtin_amdgcn_wmma_f32_16x16x4_f32(false, a0, false, b1v, (short)0, c01, false, false);
            c02 = __builtin_amdgcn_wmma_f32_16x16x4_f32(false, a0, false, b2v, (short)0, c02, false, false);
            c03 = __builtin_amdgcn_wmma_f32_16x16x4_f32(false, a0, false, b3v, (short)0, c03, false, false);
            c10 = __builtin_amdgcn_wmma_f32_16x16x4_f32(false, a1, false, b0,  (short)0, c10, false, false);
            c11 = __builtin_amdgcn_wmma_f32_16x16x4_f32(false, a1, false, b1v, (short)0, c11, false, false);
            c12 = __builtin_amdgcn_wmma_f32_16x16x4_f32(false, a1, false, b2v, (short)0, c12, false, false);
            c13 = __builtin_amdgcn_wmma_f32_16x16x4_f32(false, a1, false, b3v, (short)0, c13, false, false);
        }
        __syncthreads();
    }

    // bias + softmax epilogue (per row: 4 regs * 16 lanes of a half-wave = 64)
    const float bb0 = b2[am], bb1 = b2[am + 16], bb2 = b2[am + 32], bb3 = b2[am + 48];
    const int rhalf = (lane >> 4) << 3;
    #pragma unroll
    for (int mi = 0; mi < 2; ++mi) {
        const int rowb = m0 + (wave << 5) + (mi << 4) + rhalf;
        #pragma unroll
        for (int i = 0; i < 8; ++i) {
            float l0 = (mi ? c10[i] : c00[i]) + bb0;
            float l1 = (mi ? c11[i] : c01[i]) + bb1;
            float l2 = (mi ? c12[i] : c02[i]) + bb2;
            float l3 = (mi ? c13[i] : c03[i]) + bb3;
            float mx = fmaxf(fmaxf(l0, l1), fmaxf(l2, l3));
            #pragma unroll
            for (int off = 8; off >= 1; off >>= 1)      // stays within 16-lane half
                mx = fmaxf(mx, __shfl_xor(mx, off, 32));
            float e0 = __expf(l0 - mx), e1 = __expf(l1 - mx);
            float e2 = __expf(l2 - mx), e3 = __expf(l3 - mx);
            float s = e0 + e1 + e2 + e3;
            #pragma unroll
            for (int off = 8; off >= 1; off >>= 1)
                s += __shfl_xor(s, off, 32);
            const float inv = 1.0f / s;
            const size_t ro = (size_t)(rowb + i) * NUM_EXPERTS;
            probs[ro + am     ] = e0 * inv;
            probs[ro + am + 16] = e1 * inv;
            probs[ro + am + 32] = e2 * inv;
            probs[ro + am + 48] = e3 * inv;
        }
    }
}

// ---------------------------------------------------------------------------
// Kernel C: sequential load-masking scan (single wave32, 2 experts per lane)
// ---------------------------------------------------------------------------
__global__ __launch_bounds__(32) void scan_kernel(
    const float* __restrict__ probs, float* __restrict__ out)
{
    const int lane = threadIdx.x;     // 0..31, owns experts 2*lane, 2*lane+1
    float rx = 0.0f, ry = 0.0f;       // running_total_assignment

    for (int i = 0; i < BATCH; ++i) {
        const float2 p = *(const float2*)(probs + (size_t)i * NUM_EXPERTS + lane * 2);
        if (i + 8 < BATCH)
            __builtin_prefetch(probs + (size_t)(i + 8) * NUM_EXPERTS + lane * 2, 0, 3);

        const float r1x = rx + p.x, r1y = ry + p.y;

        // mean over 64 experts
        float s = r1x + r1y;
        #pragma unroll
        for (int off = 16; off >= 1; off >>= 1) s += __shfl_xor(s, off, 32);
        const float thr = s * (1.0f / NUM_EXPERTS) + MARGIN;

        const bool mxm = r1x > thr, mym = r1y > thr;
        float msx = mxm ? 0.0f : p.x;
        float msy = mym ? 0.0f : p.y;

        // norm and any(mask) reduced together (independent shuffle chains)
        float nrm = msx + msy;
        float fl  = (mxm || mym) ? 1.0f : 0.0f;
        #pragma unroll
        for (int off = 16; off >= 1; off >>= 1) {
            nrm += __shfl_xor(nrm, off, 32);
            fl  += __shfl_xor(fl,  off, 32);
        }
        const bool anym = fl > 0.0f;
        const float r2x = anym ? (r1x - p.x) : r1x;    // == old rta if any overloaded
        const float r2y = anym ? (r1y - p.y) : r1y;

        const float inv = 1.0f / ((nrm == 0.0f) ? 1.0f : nrm);
        msx *= inv; msy *= inv;

        float2 o; o.x = msx; o.y = msy;
        *(float2*)(out + (size_t)i * NUM_EXPERTS + lane * 2) = o;

        rx = r2x + msx;
        ry = r2y + msy;
    }
}

// ---------------------------------------------------------------------------
extern "C" void kernel_launch(void* const* d_in, const int* in_sizes, int n_in,
                              void* d_out, int out_size, void* d_ws, size_t ws_size,
                              hipStream_t stream)
{
    const float* x  = (const float*)d_in[0];   // [8192, 2048]
    const float* W1 = (const float*)d_in[1];   // [1024, 2048]
    const float* b1 = (const float*)d_in[2];   // [1024]
    const float* W2 = (const float*)d_in[3];   // [64, 1024]
    const float* b2 = (const float*)d_in[4];   // [64]
    float* out = (float*)d_out;                // [8192, 64]

    float* h     = (float*)d_ws;                            // 8192*1024 f32 (32 MB)
    float* probs = h + (size_t)BATCH * HIDDEN_SIZE;         // 8192*64  f32 ( 2 MB)

    dim3 gridA(HIDDEN_SIZE / 64, BATCH / 256);
    gemm1_relu_kernel<<<gridA, 256, 0, stream>>>(x, W1, b1, h);

    dim3 gridB(BATCH / 256);
    gemm2_softmax_kernel<<<gridB, 256, 0, stream>>>(h, W2, b2, probs);

    scan_kernel<<<1, 32, 0, stream>>>(probs, out);
}